// ChannelSparseConnectionEinsum_35845797053222
// MI455X (gfx1250) — compile-verified
//
#include <hip/hip_runtime.h>
#include <math.h>
#include <stdint.h>

typedef __attribute__((ext_vector_type(2))) float v2f;
typedef __attribute__((ext_vector_type(8))) float v8f;
typedef __attribute__((ext_vector_type(4))) unsigned int v4u;
typedef __attribute__((ext_vector_type(8))) int v8i;
typedef __attribute__((ext_vector_type(4))) int v4i;

#define NTOK 4096      // B*L
#define C1D  1024
#define C2D  1024
#define GD   256       // bottleneck groups
#define TK   32        // top-k
#define BN_EPS 1e-5f

#define KCHUNK 32      // K depth staged per TDM transfer
#define ASTR   34      // A tile LDS row stride (32 + 2 pad dwords)
#define B0STR  80      // B tile LDS row stride, BT=0 (64 + 16 pad dwords)
#define B1STR  34      // B tile LDS row stride, BT=1 (32 + 2 pad dwords)

// ---------------------------------------------------------------------------
// TDM descriptor builders (CDNA5 D#, CDNA5 ISA).
// ---------------------------------------------------------------------------
__device__ __forceinline__ v4u tdm_g0(unsigned lds_off, unsigned long long gaddr) {
  v4u g;
  g[0] = 1u;                                   // count=1 (valid), gather off
  g[1] = lds_off;                              // lds_addr (bytes)
  g[2] = (unsigned)gaddr;                      // global_addr[31:0]
  g[3] = (unsigned)((gaddr >> 32) & 0x1ffffffULL) | (2u << 30);  // [56:32] | type=2
  return g;
}

// 2-D tile: tile_dim0 x tile_dim1 elements of 4 bytes, row stride = stride0.
// pad_int/pad_amt: LDS padding codes (interval 2^(c+1) dwords, amount c+1 dwords)
__device__ __forceinline__ v8i tdm_g1(unsigned td0, unsigned td1,
                                      unsigned tile0, unsigned tile1,
                                      unsigned long long stride0,
                                      unsigned pad_int, unsigned pad_amt) {
  v8i g;
  g[0] = (int)((2u << 16) |                    // data_size = 4 bytes
               (1u << 20) |                    // pad_enable
               (pad_int << 22) | (pad_amt << 25));
  g[1] = (int)((td0 & 0xffffu) << 16);         // tensor_dim0[15:0]
  g[2] = (int)((td0 >> 16) | ((td1 & 0xffffu) << 16));
  g[3] = (int)((td1 >> 16) | (tile0 << 16));   // tile_dim0
  g[4] = (int)tile1;                           // tile_dim1 (tile_dim2 = 0)
  g[5] = (int)(unsigned)stride0;               // tensor_dim0_stride[31:0]
  g[6] = (int)(unsigned)(stride0 >> 32);       // stride[47:32] (dim1 stride unused)
  g[7] = 0;
  return g;
}

// ---------------------------------------------------------------------------
// 1) Grouped 1x1 conv (4 in -> 1 out per group), both branches fused.
// ---------------------------------------------------------------------------
__global__ __launch_bounds__(256) void conv_bottleneck(
    const float* __restrict__ x,
    const float* __restrict__ so_w, const float* __restrict__ so_b,
    const float* __restrict__ si_w, const float* __restrict__ si_b,
    float* __restrict__ xc_so, float* __restrict__ xc_si) {
  int id = blockIdx.x * 256 + threadIdx.x;          // 0 .. NTOK*GD-1
  int n = id >> 8, g = id & (GD - 1);
  float4 xv = *(const float4*)(x + n * C1D + g * 4);
  float4 w0 = *(const float4*)(so_w + g * 4);
  float4 w1 = *(const float4*)(si_w + g * 4);
  xc_so[id] = fmaf(xv.x, w0.x, fmaf(xv.y, w0.y, fmaf(xv.z, w0.z, fmaf(xv.w, w0.w, so_b[g]))));
  xc_si[id] = fmaf(xv.x, w1.x, fmaf(xv.y, w1.y, fmaf(xv.z, w1.z, fmaf(xv.w, w1.w, si_b[g]))));
}

// ---------------------------------------------------------------------------
// 2) BatchNorm training-mode stats over N per group; fold into scale/shift.
// ---------------------------------------------------------------------------
__global__ __launch_bounds__(256) void bn_stats(
    const float* __restrict__ xc, const float* __restrict__ gamma,
    const float* __restrict__ beta, float* __restrict__ scale,
    float* __restrict__ shift) {
  __shared__ float s_sum[256], s_sq[256];
  int g = blockIdx.x, t = threadIdx.x;
  float sum = 0.f, sq = 0.f;
  for (int n = t; n < NTOK; n += 256) {
    float v = xc[n * GD + g];
    sum += v; sq += v * v;
  }
  s_sum[t] = sum; s_sq[t] = sq;
  __syncthreads();
  for (int s = 128; s > 0; s >>= 1) {
    if (t < s) { s_sum[t] += s_sum[t + s]; s_sq[t] += s_sq[t + s]; }
    __syncthreads();
  }
  if (t == 0) {
    float mean = s_sum[0] * (1.f / NTOK);
    float var  = s_sq[0] * (1.f / NTOK) - mean * mean;
    float sc   = gamma[g] * rsqrtf(var + BN_EPS);
    scale[g] = sc;
    shift[g] = beta[g] - mean * sc;
  }
}

// ---------------------------------------------------------------------------
// 3) BN-apply + exact GELU, in place.
// ---------------------------------------------------------------------------
__global__ __launch_bounds__(256) void bn_gelu(
    float* __restrict__ xc, const float* __restrict__ scale,
    const float* __restrict__ shift) {
  int id = blockIdx.x * 256 + threadIdx.x;
  int g = id & (GD - 1);
  float z = fmaf(xc[id], scale[g], shift[g]);
  xc[id] = 0.5f * z * (1.f + erff(z * 0.70710678118654752440f));
}

// ---------------------------------------------------------------------------
// 4) fp32 WMMA GEMM with TDM double-buffered LDS staging.
//    C[M,Cout] = A[M,Kd] x B (+bias)
//    BT==0: B stored [Kd,Cout]  -> full = x @ weight   (LDS tile [k][n])
//    BT==1: B stored [Cout,Kd]  -> scores = xa @ lin_w.T (LDS tile [n][k])
//    Block = 128 thr = 4 waves. Block tile 64(M) x 64(N), K chunks of 32.
//    Wave 0 drives the Tensor Data Mover; TENSORcnt + s_barrier synchronize.
// ---------------------------------------------------------------------------
template <int BT>
__global__ __launch_bounds__(128) void wmma_gemm_tdm(
    const float* __restrict__ A, const float* __restrict__ B,
    const float* __restrict__ bias, float* __restrict__ C,
    int Kd, int Cout) {
  __shared__ float sA[2][64 * ASTR];        // 2 x 8704 B
  __shared__ float sB[2][KCHUNK * B0STR];   // 2 x 10240 B (covers BT=1 size too)

  const int lane  = threadIdx.x & 31;
  const int wave  = threadIdx.x >> 5;
  const int half  = lane >> 4;              // K pair select
  const int lan16 = lane & 15;
  const int m0 = blockIdx.y * 64;
  const int n0 = blockIdx.x * 64;

  v8f acc[4];
  acc[0] = v8f{}; acc[1] = v8f{}; acc[2] = v8f{}; acc[3] = v8f{};

  // --- TDM descriptors (uniform -> SGPRs) ---
  const unsigned ldsA0 = (unsigned)(uintptr_t)&sA[0][0];
  const unsigned ldsA1 = (unsigned)(uintptr_t)&sA[1][0];
  const unsigned ldsB0 = (unsigned)(uintptr_t)&sB[0][0];
  const unsigned ldsB1 = (unsigned)(uintptr_t)&sB[1][0];

  // A tile: 64 rows x 32 k, row stride Kd; LDS rows padded 32->34 dwords
  const v8i g1A = tdm_g1(1u << 20, 1u << 20, KCHUNK, 64,
                         (unsigned long long)Kd, /*pad 32dw*/4, /*+2dw*/1);
  // B tile
  const v8i g1B = (BT == 0)
      // 32 k-rows x 64 n, row stride Cout; LDS rows padded 64->80 dwords
      ? tdm_g1(1u << 20, 1u << 20, 64, KCHUNK,
               (unsigned long long)Cout, /*pad 64dw*/5, /*+16dw*/15)
      // 64 n-rows x 32 k, row stride Kd; LDS rows padded 32->34 dwords
      : tdm_g1(1u << 20, 1u << 20, KCHUNK, 64,
               (unsigned long long)Kd, /*pad 32dw*/4, /*+2dw*/1);

  const unsigned long long baseA =
      (unsigned long long)(uintptr_t)(A + (size_t)m0 * Kd);
  const unsigned long long baseB = (BT == 0)
      ? (unsigned long long)(uintptr_t)(B + n0)
      : (unsigned long long)(uintptr_t)(B + (size_t)n0 * Kd);
  const unsigned long long stepA = (unsigned long long)KCHUNK * 4ull;
  const unsigned long long stepB = (BT == 0)
      ? (unsigned long long)KCHUNK * (unsigned long long)Cout * 4ull
      : (unsigned long long)KCHUNK * 4ull;

  const int nchunks = Kd / KCHUNK;
  const v4i z4 = {};
  const v8i z8 = {};

  // prologue: stage chunk 0
  if (wave == 0) {
    __builtin_amdgcn_tensor_load_to_lds(tdm_g0(ldsA0, baseA), g1A, z4, z4, z8, 0);
    __builtin_amdgcn_tensor_load_to_lds(tdm_g0(ldsB0, baseB), g1B, z4, z4, z8, 0);
    __builtin_amdgcn_s_wait_tensorcnt(0);
  }
  __syncthreads();

  for (int c = 0; c < nchunks; ++c) {
    const int buf = c & 1;
    // kick off DMA for chunk c+1 into the other buffer (overlaps compute)
    if (wave == 0 && c + 1 < nchunks) {
      unsigned long long ga = baseA + (unsigned long long)(c + 1) * stepA;
      unsigned long long gb = baseB + (unsigned long long)(c + 1) * stepB;
      __builtin_amdgcn_tensor_load_to_lds(
          tdm_g0(buf ? ldsA0 : ldsA1, ga), g1A, z4, z4, z8, 0);
      __builtin_amdgcn_tensor_load_to_lds(
          tdm_g0(buf ? ldsB0 : ldsB1, gb), g1B, z4, z4, z8, 0);
    }

    const float* Ab = &sA[buf][0];
    const float* Bb = &sB[buf][0];
#pragma unroll
    for (int kk = 0; kk < KCHUNK; kk += 4) {
      v2f a = *(const v2f*)(Ab + (wave * 16 + lan16) * ASTR + kk + 2 * half);
      v2f b[4];
      if (BT == 0) {
#pragma unroll
        for (int nt = 0; nt < 4; ++nt) {
          b[nt][0] = Bb[(kk + 2 * half) * B0STR + nt * 16 + lan16];
          b[nt][1] = Bb[(kk + 2 * half + 1) * B0STR + nt * 16 + lan16];
        }
      } else {
#pragma unroll
        for (int nt = 0; nt < 4; ++nt)
          b[nt] = *(const v2f*)(Bb + (nt * 16 + lan16) * B1STR + kk + 2 * half);
      }
#pragma unroll
      for (int nt = 0; nt < 4; ++nt)
        acc[nt] = __builtin_amdgcn_wmma_f32_16x16x4_f32(
            false, a, false, b[nt], (short)0, acc[nt], false, false);
    }

    if (wave == 0) __builtin_amdgcn_s_wait_tensorcnt(0);
    __syncthreads();
  }

  // D layout: VGPR v, lane -> row M = v + 8*half, col N = lan16
#pragma unroll
  for (int nt = 0; nt < 4; ++nt) {
    int col = n0 + 16 * nt + lan16;
    float bv = bias ? bias[col] : 0.f;
#pragma unroll
    for (int v = 0; v < 8; ++v) {
      int row = m0 + wave * 16 + v + 8 * half;
      C[(size_t)row * Cout + col] = acc[nt][v] + bv;
    }
  }
}

// ---------------------------------------------------------------------------
// 5) Row softmax (1024) + iterative top-32 (values already softmaxed).
// ---------------------------------------------------------------------------
__global__ __launch_bounds__(256) void softmax_topk(
    const float* __restrict__ scores, float* __restrict__ vals,
    int* __restrict__ idxs) {
  __shared__ float p[1024];
  __shared__ float red[256];
  __shared__ int   redi[256];
  int n = blockIdx.x, t = threadIdx.x;
  const float* row = scores + (size_t)n * 1024;

  float lv[4];
  float lm = -INFINITY;
#pragma unroll
  for (int j = 0; j < 4; ++j) { lv[j] = row[t + 256 * j]; lm = fmaxf(lm, lv[j]); }
  red[t] = lm; __syncthreads();
  for (int s = 128; s > 0; s >>= 1) {
    if (t < s) red[t] = fmaxf(red[t], red[t + s]);
    __syncthreads();
  }
  float m = red[0]; __syncthreads();

  float ls = 0.f;
#pragma unroll
  for (int j = 0; j < 4; ++j) { float e = expf(lv[j] - m); p[t + 256 * j] = e; ls += e; }
  red[t] = ls; __syncthreads();
  for (int s = 128; s > 0; s >>= 1) {
    if (t < s) red[t] += red[t + s];
    __syncthreads();
  }
  float inv = 1.f / red[0]; __syncthreads();
#pragma unroll
  for (int j = 0; j < 4; ++j) p[t + 256 * j] *= inv;
  __syncthreads();

  for (int it = 0; it < TK; ++it) {
    float bm = -1.f; int bi = 0;
#pragma unroll
    for (int j = 0; j < 4; ++j) {
      int i = t + 256 * j;
      float pv = p[i];
      if (pv > bm) { bm = pv; bi = i; }
    }
    red[t] = bm; redi[t] = bi; __syncthreads();
    for (int s = 128; s > 0; s >>= 1) {
      if (t < s) {
        bool take = (red[t + s] > red[t]) ||
                    (red[t + s] == red[t] && redi[t + s] < redi[t]);
        if (take) { red[t] = red[t + s]; redi[t] = redi[t + s]; }
      }
      __syncthreads();
    }
    if (t == 0) {
      vals[n * TK + it] = red[0];
      idxs[n * TK + it] = redi[0];
      p[redi[0]] = -2.f;          // probs >= 0, so this removes it
    }
    __syncthreads();
  }
}

// ---------------------------------------------------------------------------
// 6) Final assembly: in_branch (32-nonzero sparse row @ W) + bias,
//    then scatter of out-branch top-k of `full` (unique indices per row).
// ---------------------------------------------------------------------------
__global__ __launch_bounds__(256) void assemble(
    const float* __restrict__ x, const float* __restrict__ W,
    const float* __restrict__ bias, const float* __restrict__ full,
    const float* __restrict__ so_val, const int* __restrict__ so_idx,
    const float* __restrict__ si_val, const int* __restrict__ si_idx,
    float* __restrict__ out) {
  __shared__ float rowbuf[1024];
  __shared__ float gate[TK];
  __shared__ int   gidx[TK];
  int n = blockIdx.x, t = threadIdx.x;

  if (t < TK) {
    int gi = si_idx[n * TK + t];
    gidx[t] = gi;
    gate[t] = si_val[n * TK + t] * x[(size_t)n * C1D + gi];
  }
  __syncthreads();

#pragma unroll
  for (int j = 0; j < 4; ++j) {
    int o = t + 256 * j;
    float acc = bias[o];
#pragma unroll
    for (int k = 0; k < TK; ++k)
      acc = fmaf(gate[k], W[(size_t)gidx[k] * C2D + o], acc);
    rowbuf[o] = acc;
  }
  __syncthreads();

  if (t < TK) {   // top-k indices of one row are unique -> no race
    int oi = so_idx[n * TK + t];
    rowbuf[oi] += full[(size_t)n * C2D + oi] * so_val[n * TK + t];
  }
  __syncthreads();

#pragma unroll
  for (int j = 0; j < 4; ++j) {
    int o = t + 256 * j;
    out[(size_t)n * C2D + o] = rowbuf[o];
  }
}

// ---------------------------------------------------------------------------
extern "C" void kernel_launch(void* const* d_in, const int* in_sizes, int n_in,
                              void* d_out, int out_size, void* d_ws, size_t ws_size,
                              hipStream_t stream) {
  const float* x         = (const float*)d_in[0];
  const float* weight    = (const float*)d_in[1];
  const float* bias      = (const float*)d_in[2];
  const float* so_conv_w = (const float*)d_in[3];
  const float* so_conv_b = (const float*)d_in[4];
  const float* so_gamma  = (const float*)d_in[5];
  const float* so_beta   = (const float*)d_in[6];
  const float* so_lin_w  = (const float*)d_in[7];
  const float* so_lin_b  = (const float*)d_in[8];
  const float* si_conv_w = (const float*)d_in[9];
  const float* si_conv_b = (const float*)d_in[10];
  const float* si_gamma  = (const float*)d_in[11];
  const float* si_beta   = (const float*)d_in[12];
  const float* si_lin_w  = (const float*)d_in[13];
  const float* si_lin_b  = (const float*)d_in[14];
  float* out = (float*)d_out;

  // --- workspace carve-up (~42 MB total) ---
  char* ws = (char*)d_ws;
  float* xc_so  = (float*)ws;  ws += (size_t)NTOK * GD * 4;     // 4 MB (in-place -> xa_so)
  float* xc_si  = (float*)ws;  ws += (size_t)NTOK * GD * 4;     // 4 MB (in-place -> xa_si)
  float* scores = (float*)ws;  ws += (size_t)NTOK * C2D * 4;    // 16 MB (reused)
  float* full   = (float*)ws;  ws += (size_t)NTOK * C2D * 4;    // 16 MB
  float* so_val = (float*)ws;  ws += (size_t)NTOK * TK * 4;
  float* si_val = (float*)ws;  ws += (size_t)NTOK * TK * 4;
  int*   so_idx = (int*)ws;    ws += (size_t)NTOK * TK * 4;
  int*   si_idx = (int*)ws;    ws += (size_t)NTOK * TK * 4;
  float* sc_so  = (float*)ws;  ws += GD * 4;
  float* sh_so  = (float*)ws;  ws += GD * 4;
  float* sc_si  = (float*)ws;  ws += GD * 4;
  float* sh_si  = (float*)ws;  ws += GD * 4;

  conv_bottleneck<<<NTOK * GD / 256, 256, 0, stream>>>(
      x, so_conv_w, so_conv_b, si_conv_w, si_conv_b, xc_so, xc_si);

  bn_stats<<<GD, 256, 0, stream>>>(xc_so, so_gamma, so_beta, sc_so, sh_so);
  bn_stats<<<GD, 256, 0, stream>>>(xc_si, si_gamma, si_beta, sc_si, sh_si);

  bn_gelu<<<NTOK * GD / 256, 256, 0, stream>>>(xc_so, sc_so, sh_so);
  bn_gelu<<<NTOK * GD / 256, 256, 0, stream>>>(xc_si, sc_si, sh_si);

  dim3 gemm_grid(C2D / 64, NTOK / 64);

  wmma_gemm_tdm<1><<<gemm_grid, 128, 0, stream>>>(xc_so, so_lin_w, so_lin_b,
                                                  scores, GD, C2D);
  softmax_topk<<<NTOK, 256, 0, stream>>>(scores, so_val, so_idx);

  wmma_gemm_tdm<1><<<gemm_grid, 128, 0, stream>>>(xc_si, si_lin_w, si_lin_b,
                                                  scores, GD, C1D);
  softmax_topk<<<NTOK, 256, 0, stream>>>(scores, si_val, si_idx);

  wmma_gemm_tdm<0><<<gemm_grid, 128, 0, stream>>>(x, weight, nullptr,
                                                  full, C1D, C2D);

  assemble<<<NTOK, 256, 0, stream>>>(x, weight, bias, full,
                                     so_val, so_idx, si_val, si_idx, out);
}